// RTF_58660663328946
// MI455X (gfx1250) — compile-verified
//
#include <hip/hip_runtime.h>
#include <hip/hip_bf16.h>

// Problem constants (from reference): B=4, L=8192, D=1024, N(coef)=64.
#define LSEQ   8192
#define DCH    1024
#define NB     4
#define ORD    64      // IIR order (64 a-taps, 64 b-taps)
#define CHUNK  64      // time-chunk length (== ORD so state = one chunk)
#define KDIM   192     // [u_hist(64) | w_hist(64) | u_chunk(64)]
#define KPAD   196     // padded LDS row stride (196 % 64 == 4 -> bank-conflict-free)
#define NCHUNK (LSEQ / CHUNK)   // 128 sequential chunk steps

typedef __attribute__((ext_vector_type(2))) float v2f;
typedef __attribute__((ext_vector_type(8))) float v8f;

// ---- gfx1250 async global->LDS path (guarded; falls back to sync loads) ----
#if defined(__has_builtin)
#if __has_builtin(__builtin_amdgcn_global_load_async_to_lds_b32) && \
    __has_builtin(__builtin_amdgcn_global_load_async_to_lds_b128) && \
    __has_builtin(__builtin_amdgcn_s_wait_asynccnt)
#define HAVE_ASYNC_LDS 1
#endif
#endif

#ifdef HAVE_ASYNC_LDS
// Parameter types per hipcc diagnostics (round 2):
//   b32 : (__device__ int*, ..., int, int)
//   b128: (__device__ int __attribute__((vector_size(16)))*, ..., int, int)
typedef int v4i_g __attribute__((vector_size(16)));
typedef __attribute__((address_space(1))) int    g_i32_t;
typedef __attribute__((address_space(3))) int    l_i32_t;
typedef __attribute__((address_space(1))) v4i_g  g_v4i_t;
typedef __attribute__((address_space(3))) v4i_g  l_v4i_t;

__device__ __forceinline__ void async_g2l_b32(const float* g, float* l) {
  __builtin_amdgcn_global_load_async_to_lds_b32(
      (g_i32_t*)(void*)g, (l_i32_t*)(void*)l, 0, 0);
}
__device__ __forceinline__ void async_g2l_b128(const float* g, float* l) {
  __builtin_amdgcn_global_load_async_to_lds_b128(
      (g_v4i_t*)(void*)g, (l_v4i_t*)(void*)l, 0, 0);
}
__device__ __forceinline__ void async_wait() {
  __builtin_amdgcn_s_wait_asynccnt(0);
}
#endif

// ---------------------------------------------------------------------------
// Phase A: per-channel exact chunk-transfer matrix M_d (64 x 192).
//   w_chunk(64) = M_d @ [u_hist(64); w_hist(64); u_chunk(64)]
// f = impulse response of 1/A (f[0]=1, f[n] = -sum a_i f[n-i]).
//   u_hist col (impulse at t0-q):  col[j] =  sum_m f[j-m] * b[m+q]
//   w_hist col (impulse at t0-q):  col[j] = -sum_m f[j-m] * a[m+q]
//   u_chunk Toeplitz:              col[j] = k'[j-c], k'[n] = sum_i b[i] f[n-i]
// ---------------------------------------------------------------------------
__global__ void rtf_build_mats(const float* __restrict__ ab,
                               float* __restrict__ M) {
  const int d = blockIdx.x;     // channel
  const int j = threadIdx.x;    // output row within chunk, 0..63
  __shared__ float a[ORD + 1];
  __shared__ float b[ORD + 1];
  __shared__ float f[CHUNK];

  a[j + 1] = ab[(size_t)d * ORD + j];
  b[j + 1] = ab[(size_t)(DCH + d) * ORD + j];
  if (j == 0) { a[0] = 1.0f; b[0] = 0.0f; }
  __syncthreads();

  if (j == 0) {               // sequential 64-step recurrence for f = 1/A
    f[0] = 1.0f;
    for (int n = 1; n < CHUNK; ++n) {
      float s = 0.0f;
      const int imax = (n < ORD) ? n : ORD;
      for (int i = 1; i <= imax; ++i) s += a[i] * f[n - i];
      f[n] = -s;
    }
  }
  __syncthreads();

  float* Mrow = M + ((size_t)d * CHUNK + j) * KDIM;

  for (int c = 0; c < ORD; ++c) {               // u-history block
    const int q = ORD - c;
    float s = 0.0f;
    for (int mm = 0; mm <= j && (mm + q) <= ORD; ++mm) s += f[j - mm] * b[mm + q];
    Mrow[c] = s;
  }
  for (int c = 0; c < ORD; ++c) {               // w-history block
    const int q = ORD - c;
    float s = 0.0f;
    for (int mm = 0; mm <= j && (mm + q) <= ORD; ++mm) s += f[j - mm] * a[mm + q];
    Mrow[ORD + c] = -s;
  }
  for (int c = 0; c < CHUNK; ++c) {             // u-chunk Toeplitz of k'
    const int n = j - c;
    float s = 0.0f;
    if (n >= 1) {
      const int imax = (n < ORD) ? n : ORD;
      for (int i = 1; i <= imax; ++i) s += b[i] * f[n - i];
    }
    Mrow[128 + c] = s;
  }
}

// ---------------------------------------------------------------------------
// Phase B: sequential chunk scan, one block (128 thr = 4 waves) per channel.
// Wave w computes rows [16w,16w+16) of w_chunk = M_d @ X via
// V_WMMA_F32_16X16X4_F32, 48 k-steps. Batches occupy WMMA columns 0..3;
// columns 4..15 are permanently-zero LDS columns so all fragment loads are
// unconditional (no EXEC-mask churn). Row strides padded to 196 dwords so
// the 16 lanes of a fragment load hit 16 distinct LDS banks.
// ---------------------------------------------------------------------------
__global__ void __launch_bounds__(128)
rtf_chunk_scan(const float* __restrict__ u,
               const float* __restrict__ h0,
               const float* __restrict__ M,
               float* __restrict__ y) {
  const int d    = blockIdx.x;
  const int tid  = threadIdx.x;      // 0..127
  const int wv   = tid >> 5;         // wave 0..3
  const int lane = tid & 31;

  __shared__ float Ms[CHUNK * KPAD]; // ~49 KB: per-channel transfer matrix
  __shared__ float X[16][KPAD];      // 16 WMMA columns x padded K (~12.5 KB)

  // Load M_d (64x192, contiguous in global) into padded LDS rows.
  const float* Mg = M + (size_t)d * CHUNK * KDIM;
#ifdef HAVE_ASYNC_LDS
  // 3072 16-byte packets via the gfx1250 async global->LDS engine.
  for (int i4 = tid; i4 < CHUNK * KDIM / 4; i4 += 128) {
    const int r = i4 / (KDIM / 4), c4 = i4 % (KDIM / 4);
    async_g2l_b128(Mg + r * KDIM + 4 * c4, &Ms[r * KPAD + 4 * c4]);
  }
#else
  for (int i = tid; i < CHUNK * KDIM; i += 128) {
    const int r = i / KDIM, c = i % KDIM;
    Ms[r * KPAD + c] = Mg[i];
  }
#endif
  // Zero all of X: cols 0..3 get live data each chunk, cols 4..15 stay zero.
  for (int i = tid; i < 16 * KPAD; i += 128) X[0][i] = 0.0f;
  const float h = h0[d];
#ifdef HAVE_ASYNC_LDS
  async_wait();
#endif
  __syncthreads();

  // WMMA f32 16x16x4 lane mapping (ISA 7.12.2):
  //  A (16x4):  lanes 0-15 row r, VGPR0/1 = K 0/1; lanes 16-31 = K 2/3
  //  B (4x16):  lanes 0-15 col n, VGPR0/1 = K 0/1; lanes 16-31 = K 2/3
  //  C/D(16x16): VGPR v, lanes 0-15 -> row v, lanes 16-31 -> row v+8
  const int r0    = wv * 16;
  const int rowA  = r0 + (lane & 15);
  const int khalf = (lane >> 4) ? 2 : 0;
  const int coln  = lane & 15;
  const int hi8   = (lane >> 4) ? 8 : 0;

  for (int ci = 0; ci < NCHUNK; ++ci) {
    const int t0 = ci * CHUNK;

    // Stage u_chunk for all 4 batches into LDS columns 0..3.
    for (int e = tid; e < NB * CHUNK; e += 128) {
      const int b = e >> 6, j = e & 63;
      const float* gp = &u[((size_t)(b * LSEQ + t0 + j)) * DCH + d];
#ifdef HAVE_ASYNC_LDS
      async_g2l_b32(gp, &X[b][128 + j]);
#else
      X[b][128 + j] = *gp;
#endif
    }
    // Prefetch next chunk toward L2 (global_prefetch_b8).
    if (ci + 1 < NCHUNK) {
      for (int e = tid; e < NB * CHUNK; e += 128) {
        const int b = e >> 6, j = e & 63;
        __builtin_prefetch(&u[((size_t)(b * LSEQ + t0 + CHUNK + j)) * DCH + d], 0, 0);
      }
    }
#ifdef HAVE_ASYNC_LDS
    async_wait();
#endif
    __syncthreads();

    // w_tile(16x16) = A(16 x 192) @ X via 48 chained f32 WMMAs.
    v8f acc = {};
#pragma unroll
    for (int kk = 0; kk < KDIM / 4; ++kk) {
      const int kb = 4 * kk + khalf;
      v2f af, bf;
      af.x = Ms[rowA * KPAD + kb];
      af.y = Ms[rowA * KPAD + kb + 1];
      bf.x = X[coln][kb];
      bf.y = X[coln][kb + 1];
      acc = __builtin_amdgcn_wmma_f32_16x16x4_f32(
          /*neg_a=*/false, af, /*neg_b=*/false, bf,
          /*c_mod=*/(short)0, acc, /*reuse_a=*/false, /*reuse_b=*/false);
    }
    __syncthreads();   // everyone done reading X before state update

    // Scatter results: y = w + h0*u ; new w_hist = w_chunk.
    if (coln < NB) {
      const int b = coln;
      const int rb = r0 + hi8;
#pragma unroll
      for (int v = 0; v < 8; ++v) {
        const int j = rb + v;
        const float w = acc[v];
        X[b][ORD + j] = w;  // w_hist zone [64,128)
        y[((size_t)(b * LSEQ + t0 + j)) * DCH + d] = w + h * X[b][128 + j];
      }
    }
    // new u_hist = u_chunk (C == ORD, so the whole chunk is the history).
    for (int e = tid; e < NB * CHUNK; e += 128) {
      const int b = e >> 6, j = e & 63;
      X[b][j] = X[b][128 + j];
    }
    __syncthreads();
  }
}

// ---------------------------------------------------------------------------
// d_in: [0]=u (4*8192*1024 f32), [1]=ab (2048*64 f32), [2]=h_0 (1024 f32)
// d_out: y (4*8192*1024 f32).  d_ws: 48 MB for per-channel matrices.
// ---------------------------------------------------------------------------
extern "C" void kernel_launch(void* const* d_in, const int* in_sizes, int n_in,
                              void* d_out, int out_size, void* d_ws, size_t ws_size,
                              hipStream_t stream) {
  const float* u  = (const float*)d_in[0];
  const float* ab = (const float*)d_in[1];
  const float* h0 = (const float*)d_in[2];
  float* y = (float*)d_out;
  float* M = (float*)d_ws;   // 1024 * 64 * 192 floats = 48 MB

  rtf_build_mats<<<DCH, CHUNK, 0, stream>>>(ab, M);
  rtf_chunk_scan<<<DCH, 128, 0, stream>>>(u, h0, M, y);
}